// HS_33852932227842
// MI455X (gfx1250) — compile-verified
//
#include <hip/hip_runtime.h>

// Hierarchical-softmax word2vec loss for MI455X (gfx1250, wave32).
// Memory-bound gather workload (~235 MB @ 23.3 TB/s); dot products done on the
// matrix pipe via V_WMMA_F32_16X16X4_F32 (diagonal-of-D batched dot trick),
// keeping full f32 reference precision.

#define VOCABN  200000
#define EMB     128
#define BATCHN  16384
#define CTXW    10
#define PLEN    18

#define ROWS    16
#define RSTR    132   // padded LDS row stride (floats); 132 % 64 == 4 -> conflict-free b64 frag reads
#define WAVES   4
#define NBLK    ((BATCHN / 16) / WAVES)   // 256 blocks

typedef __attribute__((ext_vector_type(2))) float v2f;
typedef __attribute__((ext_vector_type(8))) float v8f;

__global__ __launch_bounds__(WAVES * 32)
void hs_loss_kernel(const float* __restrict__ W_u,
                    const float* __restrict__ W_v,
                    const float* __restrict__ mask,
                    const int*   __restrict__ input_ids,
                    const int*   __restrict__ target_nodes,
                    const int*   __restrict__ codes,
                    float*       __restrict__ ws)
{
    __shared__ float lds[WAVES][ROWS * RSTR];
    __shared__ float red[WAVES];

    const int lane = threadIdx.x & 31;
    const int w    = threadIdx.x >> 5;
    const int tile = blockIdx.x * WAVES + w;   // 0..1023
    const int b0   = tile * 16;
    float* buf = lds[w];

    // ---- Phase 1: context means -> LDS tile (16 rows x 128, stride 132) ----
    // Each of the 32 lanes owns a contiguous float4 slice of the 128-dim row:
    // fully coalesced 512B global loads per gathered embedding row.
    for (int r = 0; r < ROWS; ++r) {
        const int* ids = input_ids + (size_t)(b0 + r) * CTXW;
        float ax = 0.f, ay = 0.f, az = 0.f, aw = 0.f;
        #pragma unroll
        for (int t = 0; t < CTXW; ++t) {
            const float4 v = *(const float4*)(W_u + (size_t)ids[t] * EMB + lane * 4);
            ax += v.x; ay += v.y; az += v.z; aw += v.w;
        }
        float4 o;
        o.x = ax * 0.1f; o.y = ay * 0.1f; o.z = az * 0.1f; o.w = aw * 0.1f;
        *(float4*)(buf + r * RSTR + lane * 4) = o;
    }

    // ---- Phase 2: hoist all 32 A-matrix fragments (16x4 f32 layout) into VGPRs.
    // Lane l<16 : row M=l, K%4 in {0,1};  lane 16+l : row M=l, K%4 in {2,3}.
    v2f a[32];
    {
        const int r = lane & 15;
        const int c = (lane >> 4) * 2;
        const float* ap = buf + r * RSTR + c;
        #pragma unroll
        for (int kk = 0; kk < 32; ++kk)
            a[kk] = *(const v2f*)(ap + kk * 4);
    }
    asm volatile("s_wait_dscnt 0" ::: "memory");  // A frags resident before buf reuse

    float partial = 0.0f;

    const int    rB = lane & 15;
    const int    cB = (lane >> 4) * 2;
    const float* bp = buf + rB * RSTR + cB;

    for (int l = 0; l < PLEN; ++l) {
        // Prefetch next path position's node rows (global_prefetch_b8).
        if (lane < 16 && (l + 1) < PLEN) {
            const int id2 = target_nodes[(size_t)(b0 + lane) * PLEN + (l + 1)];
            __builtin_prefetch(W_v + (size_t)id2 * EMB, 0, 1);
        }

        // Stage the 16 node rows for this path position (coalesced 512B loads).
        for (int r = 0; r < ROWS; ++r) {
            const int id = target_nodes[(size_t)(b0 + r) * PLEN + l];
            const float4 v = *(const float4*)(W_v + (size_t)id * EMB + lane * 4);
            *(float4*)(buf + r * RSTR + lane * 4) = v;
        }
        asm volatile("s_wait_dscnt 0" ::: "memory");

        // ---- 16x16 f32 WMMA over K=128 in 32 x4 steps. B is 4x16 (K x N):
        // lane n<16 : col N=n, K%4 in {0,1};  lane 16+n : col N=n, K%4 in {2,3}.
        v8f acc = {};
        #pragma unroll
        for (int kk = 0; kk < 32; ++kk) {
            const v2f bfrag = *(const v2f*)(bp + kk * 4);
            acc = __builtin_amdgcn_wmma_f32_16x16x4_f32(
                false, a[kk], false, bfrag, (short)0, acc, false, false);
        }

        // ---- Diagonal extraction: D[i][i] lives at
        //   lane v      (v=0..7) in acc[v]   (M=v,   N=v)
        //   lane 24+v   (v=0..7) in acc[v]   (M=8+v, N=8+v)
        if (lane < 8 || lane >= 24) {
            const int sel = lane & 7;
            float sc = acc[0];
            #pragma unroll
            for (int k2 = 1; k2 < 8; ++k2) sc = (sel == k2) ? acc[k2] : sc;

            const int    i  = (lane < 8) ? lane : (lane - 16);
            const size_t e  = (size_t)(b0 + i) * PLEN + l;
            const float  m  = mask[e];
            const int    cd = codes[e];
            const float  z  = cd ? sc : -sc;
            // stable log-sigmoid: min(z,0) - log1p(exp(-|z|))
            const float  ls = fminf(z, 0.0f) - log1pf(expf(-fabsf(z)));
            partial -= m * ls;
        }
    }

    // ---- Deterministic reduction: wave -> block -> d_ws partial ----
    #pragma unroll
    for (int off = 16; off > 0; off >>= 1)
        partial += __shfl_xor(partial, off, 32);
    if (lane == 0) red[w] = partial;
    __syncthreads();
    if (threadIdx.x == 0)
        ws[blockIdx.x] = red[0] + red[1] + red[2] + red[3];
}

__global__ __launch_bounds__(NBLK)
void hs_reduce_kernel(const float* __restrict__ ws, float* __restrict__ out)
{
    __shared__ float s[NBLK];
    const int t = threadIdx.x;
    s[t] = ws[t];
    __syncthreads();
    #pragma unroll
    for (int off = NBLK / 2; off > 0; off >>= 1) {
        if (t < off) s[t] += s[t + off];
        __syncthreads();
    }
    if (t == 0) out[0] = s[0];
}

extern "C" void kernel_launch(void* const* d_in, const int* in_sizes, int n_in,
                              void* d_out, int out_size, void* d_ws, size_t ws_size,
                              hipStream_t stream)
{
    (void)in_sizes; (void)n_in; (void)out_size; (void)ws_size;
    const float* W_u          = (const float*)d_in[0];
    const float* W_v          = (const float*)d_in[1];
    const float* mask         = (const float*)d_in[2];
    const int*   input_ids    = (const int*)d_in[3];
    const int*   target_nodes = (const int*)d_in[4];
    const int*   codes        = (const int*)d_in[5];
    float*       out          = (float*)d_out;
    float*       ws           = (float*)d_ws;

    hs_loss_kernel<<<NBLK, WAVES * 32, 0, stream>>>(
        W_u, W_v, mask, input_ids, target_nodes, codes, ws);
    hs_reduce_kernel<<<1, NBLK, 0, stream>>>(ws, out);
}